// IMAP_59339268161983
// MI455X (gfx1250) — compile-verified
//
#include <hip/hip_runtime.h>
#include <stdint.h>

// Problem constants (from reference): B=64, C=128, H=64, W=64
#define BSZ 64
#define CC  128
#define HWL 4096
#define TL  64                       // columns per tile staged in LDS
#define TILES_PER_B (HWL / TL)       // 64
#define NTILES (BSZ * TILES_PER_B)   // 4096
#define EPSV 1e-5f

typedef __attribute__((ext_vector_type(2))) float v2f;
typedef __attribute__((ext_vector_type(4))) float f4;
typedef __attribute__((ext_vector_type(8))) float v8f;

// ---------------------------------------------------------------------------
// Kernel 1: fold the 1x1 conv + channel-mean into wbar[c] = mean_o W[o,c],
// compute s_sig = sigmoid(s)+eps, bias_total = mean(b)+offset, and the
// constant logdet term  B * (HW/2) * sum_c log(s_sig[c]).
// ws layout (floats): [0..127]=wbar  [128..255]=s_sig  [256]=bias_total
//                     [257]=const_logdet  [512..512+4095]=per-tile partials
// ---------------------------------------------------------------------------
__global__ void __launch_bounds__(128) imap_prep_kernel(
    const float* __restrict__ w, const float* __restrict__ bias,
    const float* __restrict__ s, const float* __restrict__ offset,
    float* __restrict__ ws) {
  __shared__ float red[CC];
  const int c = threadIdx.x;  // 0..127
  float acc = 0.0f;
  for (int o = 0; o < CC; ++o) acc += w[o * CC + c];  // w[o,c,0]
  ws[c] = acc * (1.0f / CC);
  const float ss = 1.0f / (1.0f + expf(-s[c])) + EPSV;
  ws[CC + c] = ss;
  red[c] = logf(ss);
  __syncthreads();
  if (c == 0) {
    float sl = 0.0f;
    for (int i = 0; i < CC; ++i) sl += red[i];
    ws[257] = (float)BSZ * (float)(HWL / 2) * sl;  // constant logdet per batch
    float bb = 0.0f;
    for (int i = 0; i < CC; ++i) bb += bias[i];
    ws[256] = bb * (1.0f / CC) + offset[0];
  }
}

// ---------------------------------------------------------------------------
// Kernel 2: one block per (b, 64-column tile).
//   Phase 1: async-stage x[b, 0:128, l0:l0+64] (32 KB) into LDS.
//   Phase 2: WMMA f32 16x16x4 parity-split reduction -> pool -> g (per column).
//   Phase 3: out = x * ((1-mask)*g + mask*s_sig), streamed from LDS.
// ---------------------------------------------------------------------------
__global__ void __launch_bounds__(128) imap_main_kernel(
    const float* __restrict__ x, const float* __restrict__ ws,
    float* __restrict__ out, float* __restrict__ partial) {
  __shared__ float tile[CC * TL];   // tile[c*TL + l] : 32 KB
  __shared__ float wbar_s[CC];
  __shared__ float ssig_s[CC];
  __shared__ float g_s[TL];
  __shared__ float glog_s[TL];

  const int t   = threadIdx.x;       // 0..127 (4 wave32s)
  const int bid = blockIdx.x;
  const int b   = bid / TILES_PER_B;
  const int l0  = (bid % TILES_PER_B) * TL;

  // broadcast constants into LDS (t covers all 128 channels)
  wbar_s[t] = ws[t];
  ssig_s[t] = ws[CC + t];
  const float bias_total = ws[256];

  // ---- Phase 1: async global -> LDS (16B per lane per issue) ----
  {
    const int sub  = t & 15;   // 16B chunk within a 64-float row
    const int rowg = t >> 4;   // 0..7: 8 rows per iteration
    const unsigned lds_base =
        (unsigned)(uintptr_t)(&tile[0]);  // low 32 bits of generic = LDS offset
    const float* gbase = x + (size_t)b * (CC * HWL) + l0 + sub * 4;
    for (int it = 0; it < 16; ++it) {
      const int c = it * 8 + rowg;
      const unsigned lds_addr = lds_base + (unsigned)((c * TL + sub * 4) * 4);
      const float* gptr = gbase + (size_t)c * HWL;
      asm volatile("global_load_async_to_lds_b128 %0, %1, off"
                   :
                   : "v"(lds_addr), "v"(gptr)
                   : "memory");
    }
    asm volatile("s_wait_asynccnt 0x0" ::: "memory");
  }
  __syncthreads();

  // ---- Phase 2: pool via V_WMMA_F32_16X16X4_F32 ----
  // A[m,k] = wbar[4t+k] if (k&1)==(m&1) else 0  => D row0 = even-channel dot,
  // D row1 = odd-channel dot for columns l0+w*16 .. l0+w*16+15 (wave w).
  const int lane = t & 31;
  const int wv   = t >> 5;            // wave id 0..3 -> 16 columns each
  v8f acc = {};
  {
    const int hi   = (lane >> 4) & 1;     // lane group (K 0/1 vs 2/3)
    const int po   = hi * 2;
    const int woff = po + (lane & 1);
    const int col  = wv * 16 + (lane & 15);
    for (int k = 0; k < 32; ++k) {
      const int c0 = k * 4;
      const float wl = wbar_s[c0 + woff];
      v2f a;
      a.x = (lane & 1) ? 0.0f : wl;       // A VGPR0 (K = 0 or 2)
      a.y = (lane & 1) ? wl : 0.0f;       // A VGPR1 (K = 1 or 3)
      v2f bm;
      bm.x = tile[(c0 + po) * TL + col];      // B VGPR0 (K row 0 / 2)
      bm.y = tile[(c0 + po + 1) * TL + col];  // B VGPR1 (K row 1 / 3)
      acc = __builtin_amdgcn_wmma_f32_16x16x4_f32(
          false, a, false, bm, (short)0, acc, false, false);
    }
  }
  {
    const int col = wv * 16 + (lane & 15);
    const int l   = l0 + col;
    const float poolE = acc[0];  // M=0 (or 8): even-channel sum
    const float poolO = acc[1];  // M=1 (or 9): odd-channel sum
    // mask[c,l]=(c+l)%2 active => l even picks odd channels, l odd picks even
    const float pool = ((l & 1) ? poolE : poolO) + bias_total;
    const float g = 1.0f / (1.0f + expf(-pool)) + EPSV;
    if (lane < 16) {
      g_s[col]    = g;
      glog_s[col] = logf(g);
    }
  }
  __syncthreads();

  // deterministic per-tile logdet partial (reduced in kernel 3)
  if (t == 0) {
    float sl = 0.0f;
    for (int i = 0; i < TL; ++i) sl += glog_s[i];
    partial[bid] = sl;
  }

  // ---- Phase 3: out = x * attn, 4 columns per thread per row ----
  {
    const int sub  = t & 15;      // 4-float column group
    const int rowg = t >> 4;      // 0..7
    const f4 gv = *(const f4*)&g_s[sub * 4];
    const size_t outbase = (size_t)b * (CC * HWL) + l0 + sub * 4;
    for (int it = 0; it < 16; ++it) {
      const int c = it * 8 + rowg;
      const f4 xv = *(const f4*)&tile[c * TL + sub * 4];
      const float ss = ssig_s[c];
      f4 o;
      // l parity == j parity (l0, sub*4 even). mask=1 -> s_sig, mask=0 -> g.
      if (c & 1) {  // (c+j)&1 == 1 for even j
        o.x = xv.x * ss;   o.y = xv.y * gv.y;
        o.z = xv.z * ss;   o.w = xv.w * gv.w;
      } else {
        o.x = xv.x * gv.x; o.y = xv.y * ss;
        o.z = xv.z * gv.z; o.w = xv.w * ss;
      }
      *(f4*)(out + outbase + (size_t)c * HWL) = o;
    }
  }
}

// ---------------------------------------------------------------------------
// Kernel 3: logdet[b] = const + (C/2) * sum over that batch's tile partials
// (fixed-order reduction => bitwise deterministic across replays)
// ---------------------------------------------------------------------------
__global__ void imap_logdet_kernel(const float* __restrict__ partial,
                                   const float* __restrict__ ws,
                                   float* __restrict__ logdet) {
  const int b = threadIdx.x;  // 64 threads
  float s = 0.0f;
  for (int i = 0; i < TILES_PER_B; ++i) s += partial[b * TILES_PER_B + i];
  logdet[b] = ws[257] + (float)(CC / 2) * s;
}

extern "C" void kernel_launch(void* const* d_in, const int* in_sizes, int n_in,
                              void* d_out, int out_size, void* d_ws,
                              size_t ws_size, hipStream_t stream) {
  (void)in_sizes; (void)n_in; (void)out_size; (void)ws_size;
  const float* x   = (const float*)d_in[0];  // (64,128,64,64)
  const float* w   = (const float*)d_in[1];  // (128,128,1)
  const float* bb  = (const float*)d_in[2];  // (128,)
  const float* s   = (const float*)d_in[3];  // (1,128,1)
  const float* off = (const float*)d_in[4];  // (1,)

  float* out    = (float*)d_out;                        // (B,C,H,W) flat
  float* logdet = out + (size_t)BSZ * CC * HWL;         // then (B,)
  float* ws      = (float*)d_ws;
  float* partial = ws + 512;                            // 4096 floats

  imap_prep_kernel<<<1, 128, 0, stream>>>(w, bb, s, off, ws);
  imap_main_kernel<<<NTILES, 128, 0, stream>>>(x, ws, out, partial);
  imap_logdet_kernel<<<1, 64, 0, stream>>>(partial, ws, logdet);
}